// LinearGCNEncoder_9766755631464
// MI455X (gfx1250) — compile-verified
//
#include <hip/hip_runtime.h>

#define N_NODES 100000
#define E_EDGES 3200000
#define IN_DIM  512
#define OUT_DIM 128

typedef __attribute__((ext_vector_type(2))) float v2f;
typedef __attribute__((ext_vector_type(8))) float v8f;

// ---------------- degree / normalization ----------------

__global__ void zero_f32_kernel(float* __restrict__ p, int n) {
  int i = blockIdx.x * blockDim.x + threadIdx.x;
  if (i < n) p[i] = 0.0f;
}

__global__ void deg_accum_kernel(const long long* __restrict__ ei,
                                 float* __restrict__ deg) {
  int e = blockIdx.x * blockDim.x + threadIdx.x;
  if (e < E_EDGES) {
    int col = (int)ei[(size_t)E_EDGES + e];   // destination node
    unsafeAtomicAdd(&deg[col], 1.0f);         // global_atomic_add_f32, L2-resident
  }
}

__global__ void dinv_kernel(float* __restrict__ d) {
  int i = blockIdx.x * blockDim.x + threadIdx.x;
  if (i < N_NODES) d[i] = rsqrtf(d[i] + 1.0f);  // +1 for self-loop
}

// ---------------- xw = x @ W via fp32 WMMA ----------------
// Block = 16 rows of x (one M-tile), 8 waves; wave w computes the 16x16 tile
// covering output columns [16w, 16w+16). K=512 swept in steps of 4 with
// V_WMMA_F32_16X16X4_F32.
//
// fp32 16x4 A fragment (2 VGPRs): lanes 0-15 hold M=lane, {K=k, K=k+1};
// lanes 16-31 hold M=lane-16, {K=k+2, K=k+3}  -> one aligned float2 load.
// B fragment (4x16) mirrored: lanes 0-15 -> {K=k,k+1} at N=lane, lanes 16-31
// -> {K=k+2,k+3} at N=lane-16 (strided by OUT_DIM).
// C/D (8 VGPRs): vgpr g, lanes 0-15 -> M=g, lanes 16-31 -> M=g+8, N=lane%16.
__global__ __launch_bounds__(256) void gemm_wmma_kernel(
    const float* __restrict__ x, const float* __restrict__ W,
    float* __restrict__ xw) {
  const int wave = threadIdx.x >> 5;
  const int lane = threadIdx.x & 31;
  const int half = lane >> 4;       // 0 or 1
  const int lrow = lane & 15;
  const int m0 = blockIdx.x * 16;
  const int n0 = wave * 16;

  const float* arow = x + (size_t)(m0 + lrow) * IN_DIM + 2 * half;
  const float* bcol = W + (size_t)(2 * half) * OUT_DIM + (n0 + lrow);

  v8f c = {};
#pragma unroll 4
  for (int k = 0; k < IN_DIM; k += 4) {
    v2f a = *(const v2f*)(arow + k);          // {A[m][k+2h], A[m][k+2h+1]}
    v2f b;
    b.x = bcol[(size_t)k * OUT_DIM];          // W[k+2h][n]
    b.y = bcol[(size_t)k * OUT_DIM + OUT_DIM];// W[k+2h+1][n]
    // 8 args: (neg_a, A, neg_b, B, c_mod, C, reuse_a, reuse_b)
    c = __builtin_amdgcn_wmma_f32_16x16x4_f32(false, a, false, b,
                                              (short)0, c, false, false);
  }

  float* orow = xw + (size_t)(m0 + 8 * half) * OUT_DIM + n0 + lrow;
#pragma unroll
  for (int g = 0; g < 8; ++g)
    orow[(size_t)g * OUT_DIM] = c[g];
}

// ---------------- out = xw * dinv^2 + b (self-loop + bias) ----------------
__global__ void selfloop_bias_kernel(const float* __restrict__ xw,
                                     const float* __restrict__ dinv,
                                     const float* __restrict__ b,
                                     float* __restrict__ out) {
  int i = blockIdx.x * blockDim.x + threadIdx.x;   // over N*32 float4 slots
  if (i < N_NODES * (OUT_DIM / 4)) {
    int node = i >> 5;
    int j4   = i & 31;
    float s = dinv[node];
    s = s * s;
    float4 v  = ((const float4*)xw)[i];
    float4 bb = ((const float4*)b)[j4];
    float4 o;
    o.x = fmaf(v.x, s, bb.x);
    o.y = fmaf(v.y, s, bb.y);
    o.z = fmaf(v.z, s, bb.z);
    o.w = fmaf(v.w, s, bb.w);
    ((float4*)out)[i] = o;
  }
}

// ---------------- edge scatter: one wave32 per edge ----------------
// 32 lanes x float4 == the full 128-dim row. xw and out are L2-resident
// (51.2 MB each, 192 MB L2), so gather + atomics hit L2, HBM only streams
// the edge list.
__global__ __launch_bounds__(256) void edge_scatter_kernel(
    const long long* __restrict__ ei, const float* __restrict__ xw,
    const float* __restrict__ dinv, float* __restrict__ out) {
  const int lane = threadIdx.x & 31;
  const long long w =
      (long long)blockIdx.x * (blockDim.x >> 5) + (threadIdx.x >> 5);
  if (w >= E_EDGES) return;

  int row = (int)ei[w];                       // source
  int col = (int)ei[(size_t)E_EDGES + w];     // destination
  float coef = dinv[row] * dinv[col];

  float4 v = ((const float4*)(xw + (size_t)row * OUT_DIM))[lane];
  float* dst = out + (size_t)col * OUT_DIM + lane * 4;
  unsafeAtomicAdd(dst + 0, v.x * coef);
  unsafeAtomicAdd(dst + 1, v.y * coef);
  unsafeAtomicAdd(dst + 2, v.z * coef);
  unsafeAtomicAdd(dst + 3, v.w * coef);
}

// ---------------- launcher ----------------
extern "C" void kernel_launch(void* const* d_in, const int* in_sizes, int n_in,
                              void* d_out, int out_size, void* d_ws,
                              size_t ws_size, hipStream_t stream) {
  const float*     x  = (const float*)d_in[0];       // [N, 512]
  const long long* ei = (const long long*)d_in[1];   // [2, E] int64
  const float*     W  = (const float*)d_in[2];       // [512, 128]
  const float*     b  = (const float*)d_in[3];       // [128]
  float* out = (float*)d_out;                        // [N, 128]

  // workspace: xw (N*128 f32 = 51.2 MB), then deg/dinv (N f32, in place)
  float* xw   = (float*)d_ws;
  float* dinv = xw + (size_t)N_NODES * OUT_DIM;

  const int T = 256;
  zero_f32_kernel<<<(N_NODES + T - 1) / T, T, 0, stream>>>(dinv, N_NODES);
  deg_accum_kernel<<<(E_EDGES + T - 1) / T, T, 0, stream>>>(ei, dinv);
  dinv_kernel<<<(N_NODES + T - 1) / T, T, 0, stream>>>(dinv);

  gemm_wmma_kernel<<<N_NODES / 16, 256, 0, stream>>>(x, W, xw);  // 6250 blocks

  selfloop_bias_kernel<<<(N_NODES * (OUT_DIM / 4) + T - 1) / T, T, 0, stream>>>(
      xw, dinv, b, out);

  // 8 waves (edges) per 256-thread block -> 400000 blocks
  edge_scatter_kernel<<<(E_EDGES + 7) / 8, 256, 0, stream>>>(ei, xw, dinv, out);
}